// BiFusionEncoder_43885975830854
// MI455X (gfx1250) — compile-verified
//
#include <hip/hip_runtime.h>
#include <math.h>

typedef float v2f __attribute__((ext_vector_type(2)));
typedef float v8f __attribute__((ext_vector_type(8)));
typedef unsigned int u32x4 __attribute__((ext_vector_type(4)));
typedef int i32x8 __attribute__((ext_vector_type(8)));
typedef int i32x4 __attribute__((ext_vector_type(4)));

#define NPTS 16384
#define CCH  96

static __device__ __forceinline__ v8f wmma4(v2f a, v2f b, v8f c) {
  // D(16x16) = A(16x4, fp32) * B(4x16, fp32) + C
  return __builtin_amdgcn_wmma_f32_16x16x4_f32(false, a, false, b, (short)0, c,
                                               false, false);
}

// 1-D contiguous TDM load: global -> LDS, `len` fp32 elements.
static __device__ __forceinline__ void tdm_load_1d(unsigned lds_addr,
                                                   const void* gaddr,
                                                   unsigned len) {
  unsigned long long ga = (unsigned long long)(size_t)gaddr;
  u32x4 g0;
  g0[0] = 1u;                                    // count=1, user descriptor
  g0[1] = lds_addr;                              // LDS byte address
  g0[2] = (unsigned)(ga & 0xffffffffu);          // global_addr[31:0]
  g0[3] = (unsigned)((ga >> 32) & 0x01ffffffu)   // global_addr[56:32]
          | (2u << 30);                          // type = 2 ("image")
  i32x8 g1;
  g1[0] = (int)(2u << 16);                       // data_size = 4 bytes
  g1[1] = (int)((len & 0xffffu) << 16);          // tensor_dim0[15:0]
  g1[2] = (int)(((len >> 16) & 0xffffu)          // tensor_dim0[31:16]
          | (1u << 16));                         // tensor_dim1 = 1
  g1[3] = (int)((len & 0xffffu) << 16);          // tile_dim0 = len
  g1[4] = 0;                                     // tile_dim1/2 = 0 (unused)
  g1[5] = (int)len;                              // tensor_dim0_stride
  g1[6] = 0;
  g1[7] = 0;
  i32x4 z4 = {0, 0, 0, 0};
  i32x8 z8 = {0, 0, 0, 0, 0, 0, 0, 0};
  __builtin_amdgcn_tensor_load_to_lds(g0, g1, z4, z4, z8, 0);
}

// ---------------------------------------------------------------- sq = |xyz|^2
__global__ __launch_bounds__(256) void sq_kernel(const float* __restrict__ xyz,
                                                 float* __restrict__ sq) {
  int i = blockIdx.x * 256 + threadIdx.x;
  if (i < NPTS) {
    float x = xyz[i * 3 + 0], y = xyz[i * 3 + 1], z = xyz[i * 3 + 2];
    sq[i] = x * x + y * y + z * z;
  }
}

// --------------------------------------------------- fused Gram + top-16 KNN
// Candidate chunks staged into LDS by the Tensor Data Mover (double buffered),
// Gram tiles via V_WMMA_F32_16X16X4_F32, branchless ripple top-16 per lane.
// Key per column q: sq_m - 2<x_m,x_q>  (drops per-query const sq_q: order-safe)
__global__ __launch_bounds__(128) void knn_kernel(const float* __restrict__ xyz,
                                                  const float* __restrict__ sq,
                                                  int* __restrict__ knn) {
  constexpr int CHUNK = 256;
  constexpr int NCH = NPTS / CHUNK;
  __shared__ float lxyz[2][CHUNK * 3];
  __shared__ float lsq[2][CHUNK];
  __shared__ float sd[4][16][32];
  __shared__ int   si[4][16][32];
  const int lane = threadIdx.x & 31;
  const int wave = threadIdx.x >> 5;
  const int n = lane & 15, hh = lane >> 4;
  const int q = (blockIdx.x * 4 + wave) * 16 + n;

  // B (query) operand: lanes 0-15 hold (-2x,-2y); lanes 16-31 hold (-2z, 1)
  float by_alt = -2.f * xyz[q * 3 + 1];
  v2f b;
  b.x = -2.f * xyz[q * 3 + 2 * hh];
  b.y = hh ? 1.0f : by_alt;

  float bestd[16];
  int   besti[16];
#pragma unroll
  for (int t = 0; t < 16; ++t) { bestd[t] = 3.0e38f; besti[t] = 0; }

  if (wave == 0) {
    tdm_load_1d((unsigned)(size_t)&lxyz[0][0], xyz, CHUNK * 3);
    tdm_load_1d((unsigned)(size_t)&lsq[0][0], sq, CHUNK);
  }

  for (int ch = 0; ch < NCH; ++ch) {
    const int p = ch & 1;
    if (wave == 0) __builtin_amdgcn_s_wait_tensorcnt(0);
    __syncthreads();  // buffer p valid for all waves; prior reads of p^1 done
    if (wave == 0 && ch + 1 < NCH) {
      tdm_load_1d((unsigned)(size_t)&lxyz[p ^ 1][0],
                  xyz + (size_t)(ch + 1) * CHUNK * 3, CHUNK * 3);
      tdm_load_1d((unsigned)(size_t)&lsq[p ^ 1][0],
                  sq + (size_t)(ch + 1) * CHUNK, CHUNK);
    }

    for (int sub = 0; sub < CHUNK / 16; ++sub) {
      const int base = sub * 16 + n;
      v2f a;
      a.x = lxyz[p][base * 3 + 2 * hh];  // x (hh=0) or z (hh=1)
      const float* pay = hh ? &lsq[p][base] : &lxyz[p][base * 3 + 1];
      a.y = *pay;                        // y (hh=0) or sq (hh=1)

      v8f c = {};
      c = wmma4(a, b, c);

      const int cbase = ch * CHUNK + sub * 16 + 8 * hh;
      float thr = bestd[15];
#pragma unroll
      for (int r = 0; r < 8; ++r) {
        float d = c[r];
        if (d < thr) {  // rare; short branchless body, no array shifting
          float pd = d;
          int pi = cbase + r;
#pragma unroll
          for (int k = 0; k < 16; ++k) {
            bool sw = pd < bestd[k];
            float td = bestd[k];
            int ti = besti[k];
            bestd[k] = sw ? pd : td;
            besti[k] = sw ? pi : ti;
            pd = sw ? td : pd;
            pi = sw ? ti : pi;
          }
          thr = bestd[15];
        }
      }
    }
    __syncthreads();  // all waves done reading buffer p
  }

#pragma unroll
  for (int t = 0; t < 16; ++t) {
    sd[wave][n][hh * 16 + t] = bestd[t];
    si[wave][n][hh * 16 + t] = besti[t];
  }
  __syncthreads();
  if (hh == 0) {  // merge two sorted 16-lists -> 16 nearest, sorted
    int ia = 0, ib = 0;
    for (int t = 0; t < 16; ++t) {
      float da = (ia < 16) ? sd[wave][n][ia] : 3.0e38f;
      float db = (ib < 16) ? sd[wave][n][16 + ib] : 3.0e38f;
      if (da <= db) { knn[q * 16 + t] = si[wave][n][ia];      ++ia; }
      else          { knn[q * 16 + t] = si[wave][n][16 + ib]; ++ib; }
    }
  }
}

// ------------------------------- gather means + covariance normals (1 wave/pt)
__global__ __launch_bounds__(256) void gather_kernel(
    const float* __restrict__ f2d, const float* __restrict__ f3d,
    const float* __restrict__ xyz, const int* __restrict__ knn,
    float* __restrict__ fusion, float* __restrict__ normals) {
  const int lane = threadIdx.x & 31;
  const int wave = threadIdx.x >> 5;
  const int i = blockIdx.x * 8 + wave;

  int nb[16];
#pragma unroll
  for (int t = 0; t < 16; ++t) nb[t] = knn[i * 16 + t];
#pragma unroll
  for (int t = 0; t < 16; ++t) {  // WGP-scope prefetch of gathered rows
    __builtin_prefetch(f2d + (size_t)nb[t] * CCH + lane, 0, 3);
    __builtin_prefetch(f3d + (size_t)nb[t] * CCH + lane, 0, 3);
  }

  float m2[3] = {0.f, 0.f, 0.f}, m3[3] = {0.f, 0.f, 0.f};
  for (int t = 0; t < 16; ++t) {
    const float* p2 = f2d + (size_t)nb[t] * CCH;
    const float* p3 = f3d + (size_t)nb[t] * CCH;
#pragma unroll
    for (int j = 0; j < 3; ++j) {
      int ch = lane + 32 * j;
      m2[j] += p2[ch];
      m3[j] += p3[ch];
    }
  }
#pragma unroll
  for (int j = 0; j < 3; ++j) {
    int ch = lane + 32 * j;
    fusion[i * 192 + ch]      = f2d[i * CCH + ch] + m2[j] * (1.f / 16.f);
    fusion[i * 192 + 96 + ch] = f3d[i * CCH + ch] + m3[j] * (1.f / 16.f);
  }

  if (lane == 0) {
    float px = xyz[i * 3], py = xyz[i * 3 + 1], pz = xyz[i * 3 + 2];
    float a00 = 0, a01 = 0, a02 = 0, a11 = 0, a12 = 0, a22 = 0;
#pragma unroll
    for (int t = 1; t < 9; ++t) {  // knn9[:,1:] -> neighbors 1..8
      int c0 = nb[t];
      float dx = xyz[c0 * 3] - px, dy = xyz[c0 * 3 + 1] - py,
            dz = xyz[c0 * 3 + 2] - pz;
      a00 += dx * dx; a01 += dx * dy; a02 += dx * dz;
      a11 += dy * dy; a12 += dy * dz; a22 += dz * dz;
    }
    float qv = (a00 + a11 + a22) * (1.f / 3.f);
    float p1 = a01 * a01 + a02 * a02 + a12 * a12;
    float b00 = a00 - qv, b11 = a11 - qv, b22 = a22 - qv;
    float p2s = b00 * b00 + b11 * b11 + b22 * b22 + 2.f * p1;
    float p = sqrtf(p2s * (1.f / 6.f)) + 1e-20f;
    float ip = 1.f / p;
    float c00 = b00 * ip, c01 = a01 * ip, c02 = a02 * ip;
    float c11 = b11 * ip, c12 = a12 * ip, c22 = b22 * ip;
    float detB = c00 * (c11 * c22 - c12 * c12) -
                 c01 * (c01 * c22 - c12 * c02) +
                 c02 * (c01 * c12 - c11 * c02);
    float r = fminf(fmaxf(0.5f * detB, -1.f), 1.f);
    float phi = acosf(r) * (1.f / 3.f);
    float emin = qv + 2.f * p * cosf(phi + 2.0943951023931953f);
    float r0x = a00 - emin, r0y = a01, r0z = a02;
    float r1x = a01, r1y = a11 - emin, r1z = a12;
    float r2x = a02, r2y = a12, r2z = a22 - emin;
    float v0x = r0y * r1z - r0z * r1y, v0y = r0z * r1x - r0x * r1z,
          v0z = r0x * r1y - r0y * r1x;
    float v1x = r0y * r2z - r0z * r2y, v1y = r0z * r2x - r0x * r2z,
          v1z = r0x * r2y - r0y * r2x;
    float v2x = r1y * r2z - r1z * r2y, v2y = r1z * r2x - r1x * r2z,
          v2z = r1x * r2y - r1y * r2x;
    float l0 = v0x * v0x + v0y * v0y + v0z * v0z;
    float l1 = v1x * v1x + v1y * v1y + v1z * v1z;
    float l2 = v2x * v2x + v2y * v2y + v2z * v2z;
    float vx, vy, vz;
    if (l0 >= l1 && l0 >= l2) { vx = v0x; vy = v0y; vz = v0z; }
    else if (l1 >= l2)        { vx = v1x; vy = v1y; vz = v1z; }
    else                      { vx = v2x; vy = v2y; vz = v2z; }
    float nrm = fmaxf(sqrtf(vx * vx + vy * vy + vz * vz), 1e-12f);
    normals[i * 3 + 0] = vx / nrm;
    normals[i * 3 + 1] = vy / nrm;
    normals[i * 3 + 2] = vz / nrm;
  }
}

// ---------------------------------------------- tiny geo MLP 6->32->16 (VALU)
__global__ __launch_bounds__(256) void geo_kernel(
    const float* __restrict__ xyz, const float* __restrict__ normals,
    const float* __restrict__ gew1, const float* __restrict__ geb1,
    const float* __restrict__ gew2, const float* __restrict__ geb2,
    float* __restrict__ geo) {
  int i = blockIdx.x * 256 + threadIdx.x;
  if (i >= NPTS) return;
  float in6[6];
  in6[0] = xyz[i * 3];     in6[1] = xyz[i * 3 + 1];     in6[2] = xyz[i * 3 + 2];
  in6[3] = normals[i * 3]; in6[4] = normals[i * 3 + 1]; in6[5] = normals[i * 3 + 2];
  float h1[32];
#pragma unroll
  for (int o = 0; o < 32; ++o) {
    float s = geb1[o];
#pragma unroll
    for (int k = 0; k < 6; ++k) s += gew1[o * 6 + k] * in6[k];
    h1[o] = fmaxf(s, 0.f);
  }
#pragma unroll
  for (int o = 0; o < 16; ++o) {
    float s = geb2[o];
#pragma unroll
    for (int k = 0; k < 32; ++k) s += gew2[o * 32 + k] * h1[k];
    geo[i * 16 + o] = fmaxf(s, 0.f);
  }
}

// ------------------------------------------------ WMMA gate MLPs (fp32 K=4)
template <int SRC>
static __device__ __forceinline__ v2f src_pair(int row, int k,
    const float* __restrict__ f2d, const float* __restrict__ f3d,
    const float* __restrict__ fus, const float* __restrict__ geo) {
  // k is even and the (k,k+1) pair never straddles a source boundary.
  const float* p;
  if constexpr (SRC == 0) {
    p = (k < 96) ? (f2d + (size_t)row * 96 + k)
                 : (f3d + (size_t)row * 96 + (k - 96));
  } else if constexpr (SRC == 1) {
    p = fus + (size_t)row * 192 + k;
  } else {
    if (k < 96)       p = f2d + (size_t)row * 96 + k;
    else if (k < 192) p = f3d + (size_t)row * 96 + (k - 96);
    else              p = geo + (size_t)row * 16 + (k - 192);
  }
  return *(const v2f*)p;
}

template <int IN, int H1, int H2, int SRC>
__global__ __launch_bounds__(256) void gate_kernel(
    const float* __restrict__ f2d, const float* __restrict__ f3d,
    const float* __restrict__ fus, const float* __restrict__ geo,
    const float* __restrict__ W1, const float* __restrict__ B1,
    const float* __restrict__ W2, const float* __restrict__ B2,
    const float* __restrict__ Wf, const float* __restrict__ Bf,
    float* __restrict__ gate_out) {
  __shared__ float hid[8][16][64];
  const int lane = threadIdx.x & 31;
  const int wave = threadIdx.x >> 5;
  const int n = lane & 15, hh = lane >> 4;
  const int rows = (blockIdx.x * 8 + wave) * 16;
  const int m = rows + n;
  constexpr int OT1 = H1 / 16;
  constexpr int FD = (H2 > 0) ? H2 : H1;

  // ---- layer 1: [16 x IN] @ W1^T -> [16 x H1]
  v8f acc1[OT1];
#pragma unroll
  for (int ot = 0; ot < OT1; ++ot) {
    float bv = B1[ot * 16 + n];
#pragma unroll
    for (int r = 0; r < 8; ++r) acc1[ot][r] = bv;
  }
  for (int kk = 0; kk < IN; kk += 4) {
    const int k0 = kk + 2 * hh;
    v2f a = src_pair<SRC>(m, k0, f2d, f3d, fus, geo);
#pragma unroll
    for (int ot = 0; ot < OT1; ++ot) {
      v2f b = *(const v2f*)(W1 + (size_t)(ot * 16 + n) * IN + k0);
      acc1[ot] = wmma4(a, b, acc1[ot]);
    }
  }
#pragma unroll
  for (int ot = 0; ot < OT1; ++ot)
#pragma unroll
    for (int r = 0; r < 8; ++r)
      hid[wave][r + 8 * hh][ot * 16 + n] = fmaxf(acc1[ot][r], 0.f);
  asm volatile("s_wait_dscnt 0x0" ::: "memory");

  // ---- layer 2 (optional): [16 x H1] @ W2^T -> [16 x H2]
  if constexpr (H2 > 0) {
    constexpr int OT2 = H2 / 16;
    v8f acc2[OT2];
#pragma unroll
    for (int ot = 0; ot < OT2; ++ot) {
      float bv = B2[ot * 16 + n];
#pragma unroll
      for (int r = 0; r < 8; ++r) acc2[ot][r] = bv;
    }
#pragma unroll
    for (int kk = 0; kk < H1; kk += 4) {
      const int k0 = kk + 2 * hh;
      v2f a = *(const v2f*)&hid[wave][n][k0];
#pragma unroll
      for (int ot = 0; ot < OT2; ++ot) {
        v2f b = *(const v2f*)(W2 + (size_t)(ot * 16 + n) * H1 + k0);
        acc2[ot] = wmma4(a, b, acc2[ot]);
      }
    }
#pragma unroll
    for (int ot = 0; ot < OT2; ++ot)
#pragma unroll
      for (int r = 0; r < 8; ++r)
        hid[wave][r + 8 * hh][ot * 16 + n] = fmaxf(acc2[ot][r], 0.f);
    asm volatile("s_wait_dscnt 0x0" ::: "memory");
  }

  // ---- head: out=1, sigmoid
  if (hh == 0) {
    float s = Bf[0];
    for (int k = 0; k < FD; ++k) s += Wf[k] * hid[wave][n][k];
    gate_out[rows + n] = 1.f / (1.f + __expf(-s));
  }
}

// ------------------------------------------------------- final gate + conf
__global__ __launch_bounds__(256) void final_kernel(
    const float* __restrict__ bg, const float* __restrict__ sg,
    const float* __restrict__ gg, const float* __restrict__ conf_raw,
    const int* __restrict__ valid,
    const float* __restrict__ gfw1, const float* __restrict__ gfb1,
    const float* __restrict__ gfw2, const float* __restrict__ gfb2,
    float* __restrict__ fg_out, float* __restrict__ conf_out) {
  int i = blockIdx.x * 256 + threadIdx.x;
  if (i >= NPTS) return;
  float g0 = bg[i], g1 = sg[i], g2 = gg[i];
  float s = gfb2[0];
#pragma unroll
  for (int o = 0; o < 16; ++o) {
    float h = gfw1[o * 3] * g0 + gfw1[o * 3 + 1] * g1 + gfw1[o * 3 + 2] * g2 +
              gfb1[o];
    s += gfw2[o] * fmaxf(h, 0.f);
  }
  float fg = 1.f / (1.f + __expf(-s));
  float vw = (float)valid[i];
  fg = fg * vw + fg * 0.1f * (1.f - vw);
  fg_out[i] = fg;
  conf_out[i] = conf_raw[i] * (vw * 0.9f + 0.1f);
}

__global__ __launch_bounds__(256) void fuse_kernel(
    const float* __restrict__ f2d, const float* __restrict__ f3d,
    const float* __restrict__ fg, float* __restrict__ out) {
  int idx = blockIdx.x * 256 + threadIdx.x;
  if (idx >= NPTS * CCH) return;
  int i = idx / CCH;
  float g = fg[i];
  out[idx] = g * f2d[idx] + (1.f - g) * f3d[idx];
}

// ---------------------------------------------------------------- launcher
extern "C" void kernel_launch(void* const* d_in, const int* in_sizes, int n_in,
                              void* d_out, int out_size, void* d_ws,
                              size_t ws_size, hipStream_t stream) {
  const float* f2d  = (const float*)d_in[0];
  const float* f3d  = (const float*)d_in[1];
  const float* xyz  = (const float*)d_in[2];
  const int*   valid = (const int*)d_in[3];
  const float* bgw1 = (const float*)d_in[4];
  const float* bgb1 = (const float*)d_in[5];
  const float* bgw2 = (const float*)d_in[6];
  const float* bgb2 = (const float*)d_in[7];
  const float* bgw3 = (const float*)d_in[8];
  const float* bgb3 = (const float*)d_in[9];
  const float* saw1 = (const float*)d_in[10];
  const float* sab1 = (const float*)d_in[11];
  const float* saw2 = (const float*)d_in[12];
  const float* sab2 = (const float*)d_in[13];
  const float* saw3 = (const float*)d_in[14];
  const float* sab3 = (const float*)d_in[15];
  const float* gew1 = (const float*)d_in[16];
  const float* geb1 = (const float*)d_in[17];
  const float* gew2 = (const float*)d_in[18];
  const float* geb2 = (const float*)d_in[19];
  const float* cmw1 = (const float*)d_in[20];
  const float* cmb1 = (const float*)d_in[21];
  const float* cmw2 = (const float*)d_in[22];
  const float* cmb2 = (const float*)d_in[23];
  const float* cmw3 = (const float*)d_in[24];
  const float* cmb3 = (const float*)d_in[25];
  const float* gfw1 = (const float*)d_in[26];
  const float* gfb1 = (const float*)d_in[27];
  const float* gfw2 = (const float*)d_in[28];
  const float* gfb2 = (const float*)d_in[29];
  const float* cfw1 = (const float*)d_in[30];
  const float* cfb1 = (const float*)d_in[31];
  const float* cfw2 = (const float*)d_in[32];
  const float* cfb2 = (const float*)d_in[33];

  float* ws = (float*)d_ws;
  float* sq      = ws;                       // N
  int*   knn     = (int*)(ws + NPTS);        // N*16
  float* fusion  = ws + (size_t)NPTS * 17;   // N*192
  float* normals = ws + (size_t)NPTS * 209;  // N*3
  float* geo     = ws + (size_t)NPTS * 212;  // N*16
  float* bg      = ws + (size_t)NPTS * 228;  // N
  float* sg      = ws + (size_t)NPTS * 229;  // N
  float* gg      = ws + (size_t)NPTS * 230;  // N
  float* conf    = ws + (size_t)NPTS * 231;  // N
  float* fg      = ws + (size_t)NPTS * 232;  // N

  float* out_fused = (float*)d_out;           // N*96
  float* out_conf  = out_fused + NPTS * CCH;  // N

  sq_kernel<<<NPTS / 256, 256, 0, stream>>>(xyz, sq);
  knn_kernel<<<NPTS / 64, 128, 0, stream>>>(xyz, sq, knn);
  gather_kernel<<<NPTS / 8, 256, 0, stream>>>(f2d, f3d, xyz, knn, fusion,
                                              normals);
  geo_kernel<<<NPTS / 256, 256, 0, stream>>>(xyz, normals, gew1, geb1, gew2,
                                             geb2, geo);
  gate_kernel<192, 64, 32, 0><<<NPTS / 128, 256, 0, stream>>>(
      f2d, f3d, fusion, geo, bgw1, bgb1, bgw2, bgb2, bgw3, bgb3, bg);
  gate_kernel<192, 64, 64, 1><<<NPTS / 128, 256, 0, stream>>>(
      f2d, f3d, fusion, geo, saw1, sab1, saw2, sab2, saw3, sab3, sg);
  gate_kernel<208, 64, 32, 2><<<NPTS / 128, 256, 0, stream>>>(
      f2d, f3d, fusion, geo, cmw1, cmb1, cmw2, cmb2, cmw3, cmb3, gg);
  gate_kernel<192, 32, 0, 0><<<NPTS / 128, 256, 0, stream>>>(
      f2d, f3d, fusion, geo, cfw1, cfb1, nullptr, nullptr, cfw2, cfb2, conf);
  final_kernel<<<NPTS / 256, 256, 0, stream>>>(bg, sg, gg, conf, valid, gfw1,
                                               gfb1, gfw2, gfb2, fg, out_conf);
  fuse_kernel<<<(NPTS * CCH + 255) / 256, 256, 0, stream>>>(f2d, f3d, fg,
                                                            out_fused);
}